// GCN_76527727280159
// MI455X (gfx1250) — compile-verified
//
#include <hip/hip_runtime.h>
#include <math.h>

typedef __attribute__((ext_vector_type(2))) float v2f;
typedef __attribute__((ext_vector_type(8))) float v8f;

// ---------------------------------------------------------------------------
// Utility: zero fill (graph-capture-safe replacement for hipMemsetAsync)
// ---------------------------------------------------------------------------
__global__ void k_zero(float* __restrict__ p, long n) {
    long i = (long)blockIdx.x * blockDim.x + threadIdx.x;
    if (i < n) p[i] = 0.0f;
}

// ---------------------------------------------------------------------------
// Degree accumulation over dst, then dinv = rsqrt(deg + 1)  (self-loop incl.)
// ---------------------------------------------------------------------------
__global__ void k_degree(const int* __restrict__ ei, float* __restrict__ deg, int nEdges) {
    int e = blockIdx.x * blockDim.x + threadIdx.x;
    if (e < nEdges) atomicAdd(&deg[ei[nEdges + e]], 1.0f);
}

__global__ void k_dinv(float* __restrict__ deg, int n) {
    int i = blockIdx.x * blockDim.x + threadIdx.x;
    if (i < n) deg[i] = rsqrtf(deg[i] + 1.0f);
}

// ---------------------------------------------------------------------------
// WMMA GEMM: HW = H @ W  (H is [nRows x 64], W is [64 x DOUT])
// Also initializes AGG = dinv_row^2 * HW  (the fused self-loop term).
// One wave32 per 16x16 output tile, K-loop in steps of 4 using
// V_WMMA_F32_16X16X4_F32.  EXEC must be all-1s for WMMA: the only branch is
// wave-uniform (whole-wave early return for excess waves).
// ---------------------------------------------------------------------------
template <int DOUT>
__global__ void k_gemm_selfinit(const float* __restrict__ H,
                                const float* __restrict__ W,
                                const float* __restrict__ dinv,
                                float* __restrict__ HW,
                                float* __restrict__ AGG,
                                int nRows) {
    const int lane  = threadIdx.x & 31;
    const int wave  = blockIdx.x * (blockDim.x >> 5) + (threadIdx.x >> 5);
    const int tilesC   = DOUT / 16;
    const int rowTiles = nRows / 16;          // nRows is a multiple of 16
    const int rowTile  = wave / tilesC;
    const int colTile  = wave % tilesC;
    if (rowTile >= rowTiles) return;          // wave-uniform

    const int row0 = rowTile * 16;
    const int col0 = colTile * 16;
    const int hi   = lane >> 4;               // 0: lanes 0-15, 1: lanes 16-31
    const int lo   = lane & 15;

    // A fragment rows: lane lo -> M = row0 + lo ; VGPR0/1 hold K = 2*hi, 2*hi+1
    const float* __restrict__ hrow = H + (size_t)(row0 + lo) * 64;

    v8f acc = {};
    #pragma unroll
    for (int k = 0; k < 64; k += 4) {
        v2f a, b;
        a.x = hrow[k + 2 * hi];
        a.y = hrow[k + 2 * hi + 1];
        // B fragment: lane lo -> N = col0 + lo ; VGPR0/1 hold K = 2*hi, 2*hi+1
        b.x = W[(size_t)(k + 2 * hi)     * DOUT + col0 + lo];
        b.y = W[(size_t)(k + 2 * hi + 1) * DOUT + col0 + lo];
        // (neg_a, A, neg_b, B, c_mod, C, reuse_a, reuse_b)
        acc = __builtin_amdgcn_wmma_f32_16x16x4_f32(false, a, false, b,
                                                    (short)0, acc, false, false);
    }

    // C/D layout: VGPR r holds M = r (lanes 0-15) or M = 8 + r (lanes 16-31),
    // column = col0 + (lane & 15).
    const int col = col0 + lo;
    #pragma unroll
    for (int r = 0; r < 8; ++r) {
        const int row = row0 + hi * 8 + r;
        const float v = acc[r];
        HW[(size_t)row * DOUT + col] = v;
        const float di = dinv[row];
        AGG[(size_t)row * DOUT + col] = di * di * v;   // self-loop init
    }
}

// ---------------------------------------------------------------------------
// Edge scatter: AGG[dst] += dinv[src]*dinv[dst] * HW[src]
// One wave per edge; lanes cover features (float2/lane for DOUT=64).
// ---------------------------------------------------------------------------
template <int DOUT>
__global__ void k_scatter(const int* __restrict__ ei,
                          const float* __restrict__ HW,
                          const float* __restrict__ dinv,
                          float* __restrict__ AGG,
                          int nEdges) {
    const long gid  = (long)blockIdx.x * blockDim.x + threadIdx.x;
    const long e    = gid >> 5;
    if (e >= nEdges) return;
    const int lane  = (int)(gid & 31);

    const int src = ei[e];
    const int dst = ei[(long)nEdges + e];
    const float norm = dinv[src] * dinv[dst];

    if (DOUT == 64) {
        const float2 v = ((const float2*)(HW + (size_t)src * 64))[lane];
        float* dp = AGG + (size_t)dst * 64 + lane * 2;
        atomicAdd(dp,     norm * v.x);
        atomicAdd(dp + 1, norm * v.y);
    } else { // DOUT == 32
        const float v = HW[(size_t)src * 32 + lane];
        atomicAdd(&AGG[(size_t)dst * 32 + lane], norm * v);
    }
}

// ---------------------------------------------------------------------------
// h = relu(agg + b), in place (D = 64 wide)
// ---------------------------------------------------------------------------
__global__ void k_relu_bias64(float* __restrict__ agg, const float* __restrict__ b, long total) {
    long i = (long)blockIdx.x * blockDim.x + threadIdx.x;
    if (i < total) {
        float v = agg[i] + b[i & 63];
        agg[i] = v > 0.0f ? v : 0.0f;
    }
}

// ---------------------------------------------------------------------------
// Row-wise log_softmax over 32 columns; one wave per row, wave reductions
// via __shfl_xor (wave32).  Adds b3 first.
// ---------------------------------------------------------------------------
__global__ void k_logsoftmax32(float* __restrict__ out, const float* __restrict__ b, int nRows) {
    const long gid = (long)blockIdx.x * blockDim.x + threadIdx.x;
    const long row = gid >> 5;
    if (row >= nRows) return;
    const int lane = (int)(gid & 31);

    float v = out[row * 32 + lane] + b[lane];

    float m = v;
    #pragma unroll
    for (int off = 16; off > 0; off >>= 1)
        m = fmaxf(m, __shfl_xor(m, off, 32));

    float s = expf(v - m);
    #pragma unroll
    for (int off = 16; off > 0; off >>= 1)
        s += __shfl_xor(s, off, 32);

    out[row * 32 + lane] = v - m - logf(s);
}

// ---------------------------------------------------------------------------
// Host launcher
// ---------------------------------------------------------------------------
extern "C" void kernel_launch(void* const* d_in, const int* in_sizes, int n_in,
                              void* d_out, int out_size, void* d_ws, size_t ws_size,
                              hipStream_t stream) {
    const float* x  = (const float*)d_in[0];
    const int*   ei = (const int*)  d_in[1];
    const float* W1 = (const float*)d_in[2];
    const float* b1 = (const float*)d_in[3];
    const float* W2 = (const float*)d_in[4];
    const float* b2 = (const float*)d_in[5];
    const float* W3 = (const float*)d_in[6];
    const float* b3 = (const float*)d_in[7];
    float* out = (float*)d_out;

    const int N = in_sizes[0] / 64;   // 100000 (multiple of 16)
    const int E = in_sizes[1] / 2;    // 1600000

    // Workspace layout (floats): dinv[N], bufA[N*64], bufB[N*64], bufC[N*64]
    float* ws   = (float*)d_ws;
    float* dinv = ws;
    float* bufA = dinv + N;
    float* bufB = bufA + (size_t)N * 64;
    float* bufC = bufB + (size_t)N * 64;

    const int TB = 256;

    // --- degrees -> dinv ---------------------------------------------------
    k_zero<<<(N + TB - 1) / TB, TB, 0, stream>>>(dinv, (long)N);
    k_degree<<<(E + TB - 1) / TB, TB, 0, stream>>>(ei, dinv, E);
    k_dinv<<<(N + TB - 1) / TB, TB, 0, stream>>>(dinv, N);

    const long scatThreads = (long)E * 32;
    const int  scatBlocks  = (int)((scatThreads + TB - 1) / TB);

    // --- layer 1: x -> bufB ------------------------------------------------
    {
        const int waves  = (N / 16) * (64 / 16);
        const int blocks = (waves + 7) / 8;           // 8 waves / 256-thr block
        k_gemm_selfinit<64><<<blocks, TB, 0, stream>>>(x, W1, dinv, bufA, bufB, N);
        k_scatter<64><<<scatBlocks, TB, 0, stream>>>(ei, bufA, dinv, bufB, E);
        k_relu_bias64<<<(int)(((long)N * 64 + TB - 1) / TB), TB, 0, stream>>>(bufB, b1, (long)N * 64);
    }

    // --- layer 2: bufB -> bufC --------------------------------------------
    {
        const int waves  = (N / 16) * (64 / 16);
        const int blocks = (waves + 7) / 8;
        k_gemm_selfinit<64><<<blocks, TB, 0, stream>>>(bufB, W2, dinv, bufA, bufC, N);
        k_scatter<64><<<scatBlocks, TB, 0, stream>>>(ei, bufA, dinv, bufC, E);
        k_relu_bias64<<<(int)(((long)N * 64 + TB - 1) / TB), TB, 0, stream>>>(bufC, b2, (long)N * 64);
    }

    // --- layer 3: bufC -> d_out (agg lives in d_out), then log_softmax ----
    {
        const int waves  = (N / 16) * (32 / 16);
        const int blocks = (waves + 7) / 8;
        k_gemm_selfinit<32><<<blocks, TB, 0, stream>>>(bufC, W3, dinv, bufA, out, N);
        k_scatter<32><<<scatBlocks, TB, 0, stream>>>(ei, bufA, dinv, out, E);
        k_logsoftmax32<<<(int)(((long)N * 32 + TB - 1) / TB), TB, 0, stream>>>(out, b3, N);
    }
}